// MultiEmbTopModule_66030827208967
// MI455X (gfx1250) — compile-verified
//
#include <hip/hip_runtime.h>

typedef __attribute__((ext_vector_type(2))) float v2f;
typedef __attribute__((ext_vector_type(8))) float v8f;

#define SROW 66   // padded LDS row stride (floats) for 64x64 heat tile

// ---------------------------------------------------------------------------
// Stage 1: maxpool act[b][c][H0][W0] (k==stride) -> channel-pair-interleaved
//   PP [b][C/2][64][2] : PP[b][c>>1][e][c&1]  = p[b][c][e]   (e>=49 -> 0)
//   PPT[b][32][C][2]   : PPT[b][e>>1][c][e&1] = p[b][c][e]
// Pairing puts the two K-values a WMMA lane needs in one 8-byte word.
// ---------------------------------------------------------------------------
__global__ void pool_kernel(const float* __restrict__ act,
                            float* __restrict__ PP,
                            float* __restrict__ PPT,
                            int B, int C, int H0, int k) {
    int idx = blockIdx.x * blockDim.x + threadIdx.x;
    int total = B * C * 64;
    if (idx >= total) return;
    int e  = idx & 63;
    int bc = idx >> 6;
    int c  = bc % C;
    int b  = bc / C;

    float v = 0.0f;
    if (e < 49) {
        int ey = e / 7, ex = e % 7;
        const float* base = act + ((size_t)(b * C + c) * H0 + (size_t)ey * k) * H0 + ex * k;
        float m = -3.402823466e38f;
        for (int r = 0; r < k; ++r)
            for (int q = 0; q < k; ++q)
                m = fmaxf(m, base[(size_t)r * H0 + q]);
        v = m;
    }
    PP [((size_t)b * (C >> 1) + (c >> 1)) * 128 + e * 2 + (c & 1)] = v;
    PPT[((size_t)b * 32 + (e >> 1)) * ((size_t)C * 2) + (size_t)c * 2 + (e & 1)] = v;
}

// ---------------------------------------------------------------------------
// Stage 2 (fused, one block per (i,j) pair, 256 threads = 8 wave32):
//   G = Pi^T (64 x C) * Pj (C x 64)  via V_WMMA_F32_16X16X4_F32  -> LDS
//   row softmax over cols 0..48 (rows/cols >=49 zeroed), heat -> d_out
//   out = heat (64x64) * PTj (64 x C) via WMMA, rows e<49 -> d_out
// ---------------------------------------------------------------------------
__global__ void __launch_bounds__(256)
pair_kernel(const float* __restrict__ PP,   // [B][C/2][64][2]
            const float* __restrict__ PPT,  // [B][32][C][2]
            float* __restrict__ outp,       // [B][B][49][C]
            float* __restrict__ heatp,      // [B][B][49][49]
            int B, int C) {
    __shared__ float sHeat[64 * SROW];

    const int i    = blockIdx.x / B;
    const int j    = blockIdx.x % B;
    const int lane = threadIdx.x & 31;
    const int wave = threadIdx.x >> 5;           // 0..7

    const int eA    = lane & 15;                 // row (A/M) or col (B/N) index
    const int kHalf = (lane >> 4) << 1;          // lanes 16..31 hold K=2,3
    const int mBase = (lane >> 4) << 3;          // D: lanes 16..31 hold M+8

    const float* PPi  = PP  + (size_t)i * C * 64;
    const float* PPj  = PP  + (size_t)j * C * 64;
    const float* PPTj = PPT + (size_t)j * 64 * C;

    // ---- Phase 1: Gram. Wave w: nT = w&3, mT in {2*(w>>2), 2*(w>>2)+1},
    //      shared B fragment across the two M-tiles (3 b64 loads / 2 WMMA).
    {
        const int nT  = wave & 3;
        const int mT0 = (wave >> 2) << 1;
        v8f acc0 = {}, acc1 = {};
        // element offsets inside a 128-float channel-pair slab
        const int offA0 = ((mT0 * 16 + eA) << 1);
        const int offA1 = (((mT0 + 1) * 16 + eA) << 1);
        const int offB  = ((nT * 16 + eA) << 1);
        for (int c0 = 0; c0 < C; c0 += 4) {
            const size_t slab = (size_t)((c0 + kHalf) >> 1) << 7;  // *128
            v2f b  = *(const v2f*)(PPj + slab + offB);
            v2f a0 = *(const v2f*)(PPi + slab + offA0);
            v2f a1 = *(const v2f*)(PPi + slab + offA1);
            acc0 = __builtin_amdgcn_wmma_f32_16x16x4_f32(
                false, a0, false, b, (short)0, acc0, false, false);
            acc1 = __builtin_amdgcn_wmma_f32_16x16x4_f32(
                false, a1, false, b, (short)0, acc1, false, false);
        }
#pragma unroll
        for (int r = 0; r < 8; ++r) {
            sHeat[(mT0 * 16 + mBase + r) * SROW + (nT * 16 + eA)]       = acc0[r];
            sHeat[((mT0 + 1) * 16 + mBase + r) * SROW + (nT * 16 + eA)] = acc1[r];
        }
    }
    __syncthreads();

    // ---- Phase 2: row softmax (valid 49x49), zero the padding --------------
    {
        int row = threadIdx.x;
        if (row < 64) {
            float* r = sHeat + row * SROW;
            if (row < 49) {
                float mx = -3.402823466e38f;
                for (int f = 0; f < 49; ++f) mx = fmaxf(mx, r[f]);
                float s = 0.0f;
                for (int f = 0; f < 49; ++f) { float e = __expf(r[f] - mx); r[f] = e; s += e; }
                float inv = 1.0f / s;
                for (int f = 0; f < 49; ++f) r[f] *= inv;
                for (int f = 49; f < 64; ++f) r[f] = 0.0f;
            } else {
                for (int f = 0; f < 64; ++f) r[f] = 0.0f;
            }
        }
    }
    __syncthreads();

    // heat -> output [i][j][49][49]
    {
        float* hout = heatp + (size_t)(i * B + j) * 49 * 49;
        for (int idx = threadIdx.x; idx < 49 * 49; idx += 256)
            hout[idx] = sHeat[(idx / 49) * SROW + (idx % 49)];
    }

    // ---- Phase 3: out = heat(64x64) @ PTj(64 x C).
    //      Wave handles one N-tile, all 4 M-tiles: per K-step one global b64
    //      B fragment shared by 4 WMMA; A fragments are LDS b64 loads.
    {
        float* obase = outp + (size_t)(i * B + j) * 49 * C;
        const int nTiles = C >> 4;
        const size_t rowStride = (size_t)C << 1;    // floats per f-pair row in PPT
        for (int nT = wave; nT < nTiles; nT += 8) {
            v8f acc0 = {}, acc1 = {}, acc2 = {}, acc3 = {};
            const int cCol = nT * 16 + eA;
            const float* Bbase = PPTj + ((size_t)cCol << 1);
#pragma unroll 4
            for (int k0 = 0; k0 < 64; k0 += 4) {
                const int ka = k0 + kHalf;                       // even
                v2f b  = *(const v2f*)(Bbase + (size_t)(ka >> 1) * rowStride);
                v2f a0 = *(const v2f*)(&sHeat[(eA)      * SROW + ka]);
                v2f a1 = *(const v2f*)(&sHeat[(16 + eA) * SROW + ka]);
                v2f a2 = *(const v2f*)(&sHeat[(32 + eA) * SROW + ka]);
                v2f a3 = *(const v2f*)(&sHeat[(48 + eA) * SROW + ka]);
                acc0 = __builtin_amdgcn_wmma_f32_16x16x4_f32(
                    false, a0, false, b, (short)0, acc0, false, false);
                acc1 = __builtin_amdgcn_wmma_f32_16x16x4_f32(
                    false, a1, false, b, (short)0, acc1, false, false);
                acc2 = __builtin_amdgcn_wmma_f32_16x16x4_f32(
                    false, a2, false, b, (short)0, acc2, false, false);
                acc3 = __builtin_amdgcn_wmma_f32_16x16x4_f32(
                    false, a3, false, b, (short)0, acc3, false, false);
            }
#pragma unroll
            for (int r = 0; r < 8; ++r) {
                const int m = mBase + r;
                // mT = 0..3 -> e = mT*16 + m
                obase[(size_t)(m) * C + cCol] = acc0[r];          // e = m < 16 < 49
                obase[(size_t)(16 + m) * C + cCol] = acc1[r];     // e < 32
                if (32 + m < 49) obase[(size_t)(32 + m) * C + cCol] = acc2[r];
                if (48 + m < 49) obase[(size_t)(48 + m) * C + cCol] = acc3[r];
            }
        }
    }
}

// ---------------------------------------------------------------------------
extern "C" void kernel_launch(void* const* d_in, const int* in_sizes, int n_in,
                              void* d_out, int out_size, void* d_ws, size_t ws_size,
                              hipStream_t stream) {
    (void)in_sizes; (void)n_in; (void)out_size; (void)ws_size;
    const int B = 16;
    const int Cs[4] = {256, 512, 1024, 2048};
    const int Hs[4] = {56, 28, 14, 7};
    const int Ks[4] = {8, 4, 2, 1};

    float* out = (float*)d_out;
    size_t outOff[4], heatOff[4];
    size_t off = 0;
    for (int s = 0; s < 4; ++s) { outOff[s] = off; off += (size_t)B * B * 49 * Cs[s]; }
    for (int s = 0; s < 4; ++s) { heatOff[s] = off; off += (size_t)B * B * 49 * 49; }

    float* ws = (float*)d_ws;
    size_t woff = 0;
    for (int s = 0; s < 4; ++s) {
        const float* act = (const float*)d_in[s];
        const int C = Cs[s];
        float* PP  = ws + woff; woff += (size_t)B * C * 64;
        float* PPT = ws + woff; woff += (size_t)B * 64 * C;

        int poolThreads = B * C * 64;
        pool_kernel<<<(poolThreads + 255) / 256, 256, 0, stream>>>(
            act, PP, PPT, B, C, Hs[s], Ks[s]);

        pair_kernel<<<B * B, 256, 0, stream>>>(
            PP, PPT, out + outOff[s], out + heatOff[s], B, C);
    }
}